// GAT_21191368639065
// MI455X (gfx1250) — compile-verified
//
#include <hip/hip_runtime.h>
#include <math.h>

#define GAT_HEADS 5
#define GAT_INC   128
#define GAT_HID   64
#define GAT_CLS   40
#define NEG_SLOPE 0.2f

typedef __attribute__((ext_vector_type(2))) float v2f;
typedef __attribute__((ext_vector_type(8))) float v8f;

// ---------------------------------------------------------------------------
// fp32 WMMA GEMM: C[M,N] = A[M,K] @ B[K,N] (+ bias[N] if non-null)
// One wave (32 lanes) computes one 16x16 tile via V_WMMA_F32_16X16X4_F32.
// A frag (16x4): lane&15 = M row; half=lane>>4 selects K pair {0,1} / {2,3}.
// B frag (4x16): lane&15 = N col; same half/K-pair striping.
// C/D (16x16):  VGPR r, lanes0-15 -> M=r, lanes16-31 -> M=r+8; N = lane&15.
//
// N and K are template constants so all intra-chunk load/store offsets become
// immediate `offset:` fields (no per-iteration 64-bit address math).
// Out-of-range rows/cols are CLAMPED (not masked): a wrong A row / B column
// only pollutes the corresponding unstored D row/column, keeping every load
// unconditional (no EXEC save/restore).
// ---------------------------------------------------------------------------
template <int N, int K>
__global__ void gat_wmma_gemm_f32(const float* __restrict__ A,
                                  const float* __restrict__ B,
                                  const float* __restrict__ bias,
                                  float* __restrict__ C, int M) {
  constexpr int ntn = (N + 15) / 16;
  const int ntm = (M + 15) >> 4;
  const int wave = (blockIdx.x * blockDim.x + threadIdx.x) >> 5;
  if (wave >= ntm * ntn) return;            // uniform per wave
  const int lane = threadIdx.x & 31;
  const int tm = wave / ntn;
  const int tn = wave % ntn;
  const int l15  = lane & 15;
  const int half = lane >> 4;               // 0: K pair {k,k+1}, 1: {k+2,k+3}

  int row = tm * 16 + l15;                  // A row this lane feeds
  int col = tn * 16 + l15;                  // B col this lane feeds
  const int colv = col;                     // unclamped, for store guard
  row = (row < M) ? row : (M - 1);          // clamp -> unconditional loads
  col = (col < N) ? col : (N - 1);

  // Fold the per-lane K-half offset into the bases: loop offsets are constant.
  const float* __restrict__ Ap = A + (size_t)row * K + half * 2;
  const float* __restrict__ Bp = B + (size_t)(half * 2) * N + col;

  v8f acc = {0.f, 0.f, 0.f, 0.f, 0.f, 0.f, 0.f, 0.f};
#pragma unroll 8
  for (int k0 = 0; k0 < K; k0 += 4) {
    const v2f a = *(const v2f*)(Ap + k0);   // global_load_b64, imm offset
    v2f b;
    b.x = Bp[k0 * N];                       // imm offsets (constexpr stride)
    b.y = Bp[(k0 + 1) * N];
    // (neg_a, A, neg_b, B, c_mod, C, reuse_a, reuse_b)
    acc = __builtin_amdgcn_wmma_f32_16x16x4_f32(false, a, false, b,
                                                (short)0, acc, false, false);
  }

  if (colv >= N) return;
  const float bv = bias ? bias[colv] : 0.f;
  const int baseRow = tm * 16 + half * 8;
  float* __restrict__ Cp = C + (size_t)baseRow * N + colv;
  if (baseRow + 7 < M) {
    // Fast path: 8 unconditional stores, constexpr row stride -> imm offsets.
#pragma unroll
    for (int r = 0; r < 8; ++r) Cp[r * N] = acc[r] + bv;
  } else {
#pragma unroll
    for (int r = 0; r < 8; ++r) {
      if (baseRow + r < M) Cp[r * N] = acc[r] + bv;
    }
  }
}

// ---------------------------------------------------------------------------
// Attention scores: a_s[n,h] = <hp[n,h,:], att_src[h,:]>, likewise a_d.
// ---------------------------------------------------------------------------
__global__ void gat_att_scores(const float* __restrict__ hp,
                               const float* __restrict__ att_src,
                               const float* __restrict__ att_dst,
                               float* __restrict__ a_s,
                               float* __restrict__ a_d,
                               int n, int H, int C) {
  const int i = blockIdx.x * blockDim.x + threadIdx.x;
  if (i >= n * H) return;
  const int h = i % H;
  const float* v = hp + (size_t)i * C;
  float ss = 0.f, sd = 0.f;
  for (int c = 0; c < C; ++c) {
    const float x = v[c];
    ss += x * att_src[h * C + c];
    sd += x * att_dst[h * C + c];
  }
  a_s[i] = ss;
  a_d[i] = sd;
}

// Order-preserving float <-> uint mapping for atomicMax on signed floats.
__device__ __forceinline__ unsigned f2ord(float f) {
  const unsigned u = __float_as_uint(f);
  return (u & 0x80000000u) ? ~u : (u | 0x80000000u);
}
__device__ __forceinline__ float ord2f(unsigned u) {
  return __uint_as_float((u & 0x80000000u) ? (u & 0x7FFFFFFFu) : ~u);
}

__device__ __forceinline__ void edge_decode(const long long* __restrict__ ei,
                                            int e, int E, int& src, int& dst) {
  if (e < E) {
    src = (int)ei[e];
    dst = (int)ei[(size_t)E + e];
  } else {               // self-loops appended after the edge list
    src = dst = e - E;
  }
}

// Pass A: segment max over dst (atomicMax on ordered-uint).
__global__ void gat_edge_max(const long long* __restrict__ ei, int E, int n, int H,
                             const float* __restrict__ a_s,
                             const float* __restrict__ a_d,
                             unsigned* __restrict__ m) {
  const int i = blockIdx.x * blockDim.x + threadIdx.x;
  const int tot = (E + n) * H;
  if (i >= tot) return;
  const int e = i / H, h = i % H;
  int src, dst;
  edge_decode(ei, e, E, src, dst);
  float v = a_s[src * H + h] + a_d[dst * H + h];
  v = (v > 0.f) ? v : NEG_SLOPE * v;
  atomicMax(&m[dst * H + h], f2ord(v));
}

// Pass B: ex = exp(e - m[dst]); cache ex; segment sum over dst.
__global__ void gat_edge_expsum(const long long* __restrict__ ei, int E, int n, int H,
                                const float* __restrict__ a_s,
                                const float* __restrict__ a_d,
                                const unsigned* __restrict__ m,
                                float* __restrict__ s,
                                float* __restrict__ exb) {
  const int i = blockIdx.x * blockDim.x + threadIdx.x;
  const int tot = (E + n) * H;
  if (i >= tot) return;
  const int e = i / H, h = i % H;
  int src, dst;
  edge_decode(ei, e, E, src, dst);
  float v = a_s[src * H + h] + a_d[dst * H + h];
  v = (v > 0.f) ? v : NEG_SLOPE * v;
  const float ex = expf(v - ord2f(m[dst * H + h]));
  exb[i] = ex;
  atomicAdd(&s[dst * H + h], ex);
}

// Pass C: out[dst,h,c] += (ex / (s[dst,h]+1e-16)) * hp[src,h,c].
// One thread per (edge, head, channel): coalesced hp reads + coalesced atomics.
__global__ void gat_edge_msg(const long long* __restrict__ ei, int E, int n, int H, int C,
                             const float* __restrict__ exb,
                             const float* __restrict__ s,
                             const float* __restrict__ hp,
                             float* __restrict__ out) {
  const int i = blockIdx.x * blockDim.x + threadIdx.x;
  const int tot = (E + n) * H * C;
  if (i >= tot) return;
  const int c = i % C;
  const int eh = i / C;
  const int h = eh % H;
  const int e = eh / H;
  int src, dst;
  edge_decode(ei, e, E, src, dst);
  const float alpha = exb[eh] / (s[dst * H + h] + 1e-16f);
  atomicAdd(&out[(size_t)(dst * H + h) * C + c],
            alpha * hp[(size_t)(src * H + h) * C + c]);
}

// Layer-1 finalize: h1 = elu(out1 + b1), in place.  HC = H*HID.
__global__ void gat_finalize1(float* __restrict__ out1,
                              const float* __restrict__ b1,
                              int n, int HC) {
  const int i = blockIdx.x * blockDim.x + threadIdx.x;
  if (i >= n * HC) return;
  const float v = out1[i] + b1[i % HC];
  out1[i] = (v > 0.f) ? v : (expf(v) - 1.f);
}

// Layer-2 finalize: logits = mean_h(out2) + b2; then row log_softmax.
__global__ void gat_finalize2(const float* __restrict__ out2,
                              const float* __restrict__ b2,
                              float* __restrict__ out,
                              int n, int H, int C) {
  const int row = blockIdx.x * blockDim.x + threadIdx.x;
  if (row >= n) return;
  float logits[GAT_CLS];
  float mx = -3.4e38f;
  const float invH = 1.f / (float)H;
  for (int c = 0; c < C; ++c) {
    float acc = 0.f;
    for (int h = 0; h < H; ++h) acc += out2[(size_t)(row * H + h) * C + c];
    const float v = acc * invH + b2[c];
    logits[c] = v;
    mx = fmaxf(mx, v);
  }
  float se = 0.f;
  for (int c = 0; c < C; ++c) se += expf(logits[c] - mx);
  const float lse = logf(se) + mx;
  for (int c = 0; c < C; ++c) out[(size_t)row * C + c] = logits[c] - lse;
}

// ---------------------------------------------------------------------------
static inline unsigned cdiv_u(long long a, long long b) {
  return (unsigned)((a + b - 1) / b);
}

template <int N, int K>
static inline void launch_gemm(const float* A, const float* B, const float* bias,
                               float* C, int M, hipStream_t stream) {
  const long long tiles = (long long)((M + 15) / 16) * ((N + 15) / 16);
  const unsigned grid = cdiv_u(tiles * 32, 256);
  gat_wmma_gemm_f32<N, K><<<grid, 256, 0, stream>>>(A, B, bias, C, M);
}

extern "C" void kernel_launch(void* const* d_in, const int* in_sizes, int n_in,
                              void* d_out, int out_size, void* d_ws, size_t ws_size,
                              hipStream_t stream) {
  const float*     x     = (const float*)d_in[0];
  const long long* ei    = (const long long*)d_in[1];   // int64 [2,E]
  const float*     emb_W = (const float*)d_in[2];
  const float*     emb_b = (const float*)d_in[3];
  const float*     W1    = (const float*)d_in[4];
  const float*     as1   = (const float*)d_in[5];
  const float*     ad1   = (const float*)d_in[6];
  const float*     b1    = (const float*)d_in[7];
  const float*     W2    = (const float*)d_in[8];
  const float*     as2   = (const float*)d_in[9];
  const float*     ad2   = (const float*)d_in[10];
  const float*     b2    = (const float*)d_in[11];

  const int H   = GAT_HEADS;
  const int n   = in_sizes[0] / GAT_INC;
  const int E   = in_sizes[1] / 2;
  const int Et  = E + n;                    // edges + self loops
  const int HC1 = H * GAT_HID;              // 320
  const int HC2 = H * GAT_CLS;              // 200

  float* emb = (float*)d_out;                        // [n, 64]
  float* lsm = (float*)d_out + (size_t)n * GAT_HID;  // [n, 40]

  // ---- workspace layout (regions reused between layers) ----
  char* ws = (char*)d_ws;
  size_t off = 0;
  float* hp   = (float*)(ws + off); off += (size_t)n * HC1 * sizeof(float); // hp1, later hp2
  float* outb = (float*)(ws + off); off += (size_t)n * HC1 * sizeof(float); // out1/h1, later out2
  unsigned* m = (unsigned*)(ws + off); off += (size_t)n * H * sizeof(unsigned);
  float* s    = (float*)(ws + off); off += (size_t)n * H * sizeof(float);
  float* a_s  = (float*)(ws + off); off += (size_t)n * H * sizeof(float);
  float* a_d  = (float*)(ws + off); off += (size_t)n * H * sizeof(float);
  float* exb  = (float*)(ws + off); off += (size_t)Et * H * sizeof(float);

  // ---- embedding: emb = x @ emb_W + emb_b ----
  launch_gemm<GAT_HID, GAT_INC>(x, emb_W, emb_b, emb, n, stream);

  // ================= layer 1 (concat heads, HID channels) =================
  launch_gemm<GAT_HEADS * GAT_HID, GAT_HID>(emb, W1, nullptr, hp, n, stream);
  gat_att_scores<<<cdiv_u((long long)n * H, 256), 256, 0, stream>>>(
      hp, as1, ad1, a_s, a_d, n, H, GAT_HID);

  hipMemsetAsync(m, 0, (size_t)n * H * sizeof(unsigned), stream);
  hipMemsetAsync(s, 0, (size_t)n * H * sizeof(float), stream);
  hipMemsetAsync(outb, 0, (size_t)n * HC1 * sizeof(float), stream);

  gat_edge_max<<<cdiv_u((long long)Et * H, 256), 256, 0, stream>>>(
      ei, E, n, H, a_s, a_d, m);
  gat_edge_expsum<<<cdiv_u((long long)Et * H, 256), 256, 0, stream>>>(
      ei, E, n, H, a_s, a_d, m, s, exb);
  gat_edge_msg<<<cdiv_u((long long)Et * H * GAT_HID, 256), 256, 0, stream>>>(
      ei, E, n, H, GAT_HID, exb, s, hp, outb);
  gat_finalize1<<<cdiv_u((long long)n * HC1, 256), 256, 0, stream>>>(
      outb, b1, n, HC1);

  // ================= layer 2 (mean heads, CLS channels) ==================
  launch_gemm<GAT_HEADS * GAT_CLS, GAT_HEADS * GAT_HID>(outb, W2, nullptr, hp, n,
                                                        stream);  // hp -> hp2
  gat_att_scores<<<cdiv_u((long long)n * H, 256), 256, 0, stream>>>(
      hp, as2, ad2, a_s, a_d, n, H, GAT_CLS);

  hipMemsetAsync(m, 0, (size_t)n * H * sizeof(unsigned), stream);
  hipMemsetAsync(s, 0, (size_t)n * H * sizeof(float), stream);
  hipMemsetAsync(outb, 0, (size_t)n * HC2 * sizeof(float), stream); // out2 region

  gat_edge_max<<<cdiv_u((long long)Et * H, 256), 256, 0, stream>>>(
      ei, E, n, H, a_s, a_d, m);
  gat_edge_expsum<<<cdiv_u((long long)Et * H, 256), 256, 0, stream>>>(
      ei, E, n, H, a_s, a_d, m, s, exb);
  gat_edge_msg<<<cdiv_u((long long)Et * H * GAT_CLS, 256), 256, 0, stream>>>(
      ei, E, n, H, GAT_CLS, exb, s, hp, outb);

  gat_finalize2<<<cdiv_u(n, 128), 128, 0, stream>>>(outb, b2, lsm, n, H, GAT_CLS);
}